// Conv2D_57183194579431
// MI455X (gfx1250) — compile-verified
//
#include <hip/hip_runtime.h>

// ---- problem constants (from reference) ----
constexpr int IMG_H = 6144;
constexpr int IMG_W = 6144;
constexpr int FKH   = 11;
constexpr int FKW   = 11;
constexpr int OUT_H = IMG_H - FKH + 1;   // 6134
constexpr int OUT_W = IMG_W - FKW + 1;   // 6134

// ---- tiling ----
constexpr int TILE_ROWS  = 16;                 // output rows per block  (WMMA N)
constexpr int WAVES      = 8;                  // waves per block (wave32)
constexpr int BLOCK_COLS = 16 * WAVES;         // 128 output cols per block (16 per wave, WMMA M)
constexpr int KPAD       = 28;                 // K window 0..25 padded to 28 (7 wmma k-steps)
constexpr int XT_ROWS    = TILE_ROWS + FKH - 1;     // 26
constexpr int XT_COLS    = 144;                     // >= (BLOCK_COLS-16) + KPAD = 140; 36 float4/row
constexpr int XT_STRIDE  = 148;                     // floats; 592B rows: 16B aligned, conflict-free fan
constexpr int A_STRIDE   = KPAD;                    // 28

typedef __attribute__((ext_vector_type(2))) float v2f;
typedef __attribute__((ext_vector_type(8))) float v8f;

// Guard only the DEVICE pass: host pass has no amdgcn builtins and must not error.
#if defined(__HIP_DEVICE_COMPILE__)
#if !__has_builtin(__builtin_amdgcn_wmma_f32_16x16x4_f32)
#error "gfx1250 wmma f32 16x16x4 builtin not available on this toolchain (device pass)"
#endif
#endif

__device__ __forceinline__ void wait_asynccnt0() {
#if defined(__HIP_DEVICE_COMPILE__)
#if __has_builtin(__builtin_amdgcn_s_wait_asynccnt)
    __builtin_amdgcn_s_wait_asynccnt(0);
#else
    asm volatile("s_wait_asynccnt 0x0" ::: "memory");
#endif
#endif
}

__global__ __launch_bounds__(256) void
conv2d_wmma_f32(const float* __restrict__ X,
                const float* __restrict__ Wg,      // 11x11 row-major
                const float* __restrict__ bias,
                float* __restrict__ out)
{
    __shared__ float sX[XT_ROWS * XT_STRIDE];            // 26*148*4 = 15,392 B
    __shared__ float sA[FKH * 16 * A_STRIDE];            // 11*16*28*4 = 19,712 B

    const int tid  = threadIdx.x;
    const int wave = tid >> 5;
    const int lane = tid & 31;
    const int m    = lane & 15;        // WMMA M index (output col in wave tile) == N index (output row)
    const int half = (lane >> 4) & 1;  // lane half selects K pair / output col group

    const int i0 = blockIdx.y * TILE_ROWS;     // first output row of block
    const int c0 = blockIdx.x * BLOCK_COLS;    // first output col of block

    // ---- edge blocks: zero-prefill LDS tile so OOB slots read as 0 ----
    const bool edge = (i0 + XT_ROWS > IMG_H) || (c0 + XT_COLS > IMG_W);
    if (edge) {
        for (int idx = tid; idx < XT_ROWS * XT_STRIDE; idx += 256)
            sX[idx] = 0.0f;
        __syncthreads();   // ds stores complete before any async LDS writes
    }

    // ---- stage input tile via async global->LDS DMA (b128 = one float4/lane) ----
    // c0 multiple of 128 and IMG_W multiple of 128 => no float4 straddles the edge.
    for (int idx = tid; idx < XT_ROWS * (XT_COLS / 4); idx += 256) {
        const int r  = idx / (XT_COLS / 4);
        const int c4 = idx - r * (XT_COLS / 4);
        const int gr = i0 + r;
        const int gc = c0 + c4 * 4;
        if (gr < IMG_H && gc < IMG_W) {
            unsigned int       lw = (unsigned int)(size_t)&sX[r * XT_STRIDE + c4 * 4];
            unsigned long long ga = (unsigned long long)(size_t)&X[(long)gr * IMG_W + gc];
            asm volatile("global_load_async_to_lds_b128 %0, %1, off"
                         :: "v"(lw), "v"(ga) : "memory");
        }
    }

    // ---- build banded-Toeplitz A_u matrices (overlaps the async DMA) ----
    // A[u][m][t] = W[u][t-m] for 0 <= t-m < 11, else 0
    for (int idx = tid; idx < FKH * 16 * A_STRIDE; idx += 256) {
        const int u   = idx / (16 * A_STRIDE);
        const int rem = idx - u * 16 * A_STRIDE;
        const int mm  = rem / A_STRIDE;
        const int t   = rem - mm * A_STRIDE;
        const int v   = t - mm;
        sA[idx] = (v >= 0 && v < FKW) ? Wg[u * FKW + v] : 0.0f;
    }

    wait_asynccnt0();      // this wave's DMA landed in LDS
    __syncthreads();       // all waves' DMA + A-build visible

    // ---- 77 x v_wmma_f32_16x16x4_f32 per 16x16 output tile ----
    const int jw = wave * 16;                  // this wave's column offset in the block
    v8f acc = {0.f, 0.f, 0.f, 0.f, 0.f, 0.f, 0.f, 0.f};

#if defined(__HIP_DEVICE_COMPILE__)
    for (int u = 0; u < FKH; ++u) {
        // A operand: lane supplies A[u][m][4k + 2*half + {0,1}]   (16x4 f32 layout)
        const float* __restrict__ arow = &sA[(u * 16 + m) * A_STRIDE + 2 * half];
        // B operand: lane supplies X[i0 + n + u][c0 + jw + 4k + 2*half + {0,1}], n = lane&15
        const float* __restrict__ brow = &sX[(m + u) * XT_STRIDE + jw + 2 * half];
#pragma unroll
        for (int k = 0; k < KPAD / 4; ++k) {
            v2f a = *(const v2f*)(arow + 4 * k);   // 8B-aligned ds_load_b64
            v2f b = *(const v2f*)(brow + 4 * k);
            acc = __builtin_amdgcn_wmma_f32_16x16x4_f32(
                /*neg_a=*/false, a, /*neg_b=*/false, b,
                /*c_mod=*/(short)0, acc, /*reuse_a=*/false, /*reuse_b=*/false);
        }
    }
#endif

    // ---- store: D vgpr r holds (M = r + 8*half, N = lane&15) ----
    const float bv = bias[0];
    const int oi = i0 + m;                        // output row (N)
    const int jbase = c0 + jw + 8 * half;         // output col base (M)
    if (oi < OUT_H) {
#pragma unroll
        for (int r = 0; r < 8; ++r) {
            const int oj = jbase + r;
            if (oj < OUT_W)
                out[(long)oi * OUT_W + oj] = acc[r] + bv;
        }
    }
}

extern "C" void kernel_launch(void* const* d_in, const int* in_sizes, int n_in,
                              void* d_out, int out_size, void* d_ws, size_t ws_size,
                              hipStream_t stream) {
    const float* X    = (const float*)d_in[0];
    const float* Wg   = (const float*)d_in[1];
    const float* bias = (const float*)d_in[2];
    float* out = (float*)d_out;

    dim3 grid((OUT_W + BLOCK_COLS - 1) / BLOCK_COLS,   // 48
              (OUT_H + TILE_ROWS  - 1) / TILE_ROWS);   // 384
    conv2d_wmma_f32<<<grid, 256, 0, stream>>>(X, Wg, bias, out);
}